// MatchaTTS_63359357550598
// MI455X (gfx1250) — compile-verified
//
#include <hip/hip_runtime.h>

#define B_    32
#define T_    1024
#define D_    256
#define NMELS 80
#define MMEL  8192                 // T_ * MAX_DUR
#define MU_ELEMS (B_ * NMELS * MMEL)   // 20971520 floats, then B_*MMEL mask floats

typedef __attribute__((ext_vector_type(2))) float v2f;
typedef __attribute__((ext_vector_type(8))) float v8f;

// ---------------- Kernel 1: per-batch inclusive cumsum of durations ----------------
__global__ void __launch_bounds__(1024)
matcha_cumsum_k(const int* __restrict__ dur, int* __restrict__ cum) {
    __shared__ int s[T_];
    const int b = blockIdx.x;
    const int t = threadIdx.x;
    s[t] = dur[b * T_ + t];
    __syncthreads();
#pragma unroll
    for (int off = 1; off < T_; off <<= 1) {
        int v = (t >= off) ? s[t - off] : 0;
        __syncthreads();
        s[t] += v;
        __syncthreads();
    }
    cum[b * T_ + t] = s[t];
}

// ---------------- Kernel 2: searchsorted(right) + mel mask ----------------
__global__ void __launch_bounds__(256)
matcha_search_k(const int* __restrict__ cum, int* __restrict__ idxb,
                float* __restrict__ maskb) {
    const int gid = blockIdx.x * 256 + threadIdx.x;    // 0 .. B_*MMEL-1
    const int b = gid >> 13;                           // / MMEL
    const int p = gid & (MMEL - 1);
    const int* __restrict__ c = cum + b * T_;
    // count of elements <= p  (== searchsorted side='right'), branchless
    int pos = 0;
#pragma unroll
    for (int step = 512; step >= 1; step >>= 1) {
        int np = pos + step;                           // np <= 1023 always
        if (c[np - 1] <= p) pos = np;
    }
    idxb[gid]  = (pos > T_ - 1) ? (T_ - 1) : pos;      // clamp (pos <= 1023 anyway)
    maskb[gid] = (p < c[T_ - 1]) ? 1.0f : 0.0f;        // second tuple output
}

// ---------------- Kernel 3: WMMA gather-GEMM  mu_cond = (U @ W^T + b) * mask ------
// A = W_mu tile (M=mel, K=d), B = gathered encoded_text (K=d, N=frame).
// One wave: 16-mel tile x 4 consecutive 16-frame tiles (64 frames), K=256 in steps of 4.
__global__ void __launch_bounds__(128)
matcha_gemm_k(const float* __restrict__ enc,   // (B_, T_, D_)
              const float* __restrict__ Wmu,   // (NMELS, D_)
              const float* __restrict__ bmu,   // (NMELS)
              const int*   __restrict__ idxb,  // (B_, MMEL)
              const float* __restrict__ maskb, // (B_, MMEL)
              float*       __restrict__ out)   // (B_, NMELS, MMEL)
{
    const int wave = blockIdx.x * 4 + (threadIdx.x >> 5);
    const int lane = threadIdx.x & 31;
    const int hi   = lane >> 4;        // 0: K+0/1, 1: K+2/3
    const int l16  = lane & 15;

    const int melTile = wave % 5;              // 5 tiles of 16 mels (80)
    int tmp = wave / 5;
    const int fg = tmp & 127;                  // 128 groups of 64 frames (8192)
    const int b  = tmp >> 7;

    const int melbase = melTile * 16;
    const int f0 = fg * 64;
    const int kofs = 2 * hi;

    // A source row: W_mu[melbase + l16][*]
    const float* __restrict__ arow = Wmu + (size_t)(melbase + l16) * D_;

    // Gather row pointers + masks for the 4 frame sub-tiles
    const float* rowp[4];
    float maskf[4];
#pragma unroll
    for (int j = 0; j < 4; ++j) {
        const int frame = f0 + j * 16 + l16;
        const int id = idxb[b * MMEL + frame];                 // clamped, always valid
        rowp[j]  = enc + ((size_t)b * T_ + id) * D_;
        maskf[j] = maskb[b * MMEL + frame];
    }

    v8f acc[4] = {v8f{}, v8f{}, v8f{}, v8f{}};

#pragma unroll 4
    for (int kb = 0; kb < D_; kb += 4) {
        v2f a = *(const v2f*)(arow + kb + kofs);
#pragma unroll
        for (int j = 0; j < 4; ++j) {
            v2f bj = *(const v2f*)(rowp[j] + kb + kofs);
            acc[j] = __builtin_amdgcn_wmma_f32_16x16x4_f32(
                false, a, false, bj, (short)0, acc[j], false, false);
        }
    }

    // Bias per D row r: mel = melbase + 8*hi + r
    v8f bias;
#pragma unroll
    for (int r = 0; r < 8; ++r) bias[r] = bmu[melbase + 8 * hi + r];

#pragma unroll
    for (int j = 0; j < 4; ++j) {
        const int frame = f0 + j * 16 + l16;
#pragma unroll
        for (int r = 0; r < 8; ++r) {
            const int mel = melbase + 8 * hi + r;
            out[((size_t)b * NMELS + mel) * MMEL + frame] =
                (acc[j][r] + bias[r]) * maskf[j];
        }
    }
}

extern "C" void kernel_launch(void* const* d_in, const int* in_sizes, int n_in,
                              void* d_out, int out_size, void* d_ws, size_t ws_size,
                              hipStream_t stream) {
    const float* enc = (const float*)d_in[0];   // encoded_text (32,1024,256) f32
    const int*   dur = (const int*)  d_in[1];   // durations (32,1024) i32
    const float* Wmu = (const float*)d_in[2];   // (80,256) f32
    const float* bmu = (const float*)d_in[3];   // (80,) f32

    float* out   = (float*)d_out;
    float* maskb = out + MU_ELEMS;              // second tuple output: mel_mask as float

    int* cum  = (int*)d_ws;                     // B_*T_ ints
    int* idxb = cum + B_ * T_;                  // B_*MMEL ints

    matcha_cumsum_k<<<B_, 1024, 0, stream>>>(dur, cum);
    matcha_search_k<<<(B_ * MMEL) / 256, 256, 0, stream>>>(cum, idxb, maskb);

    // waves: B_ * (MMEL/64) * (NMELS/16) = 32*128*5 = 20480 ; 4 waves/block
    matcha_gemm_k<<<20480 / 4, 128, 0, stream>>>(enc, Wmu, bmu, idxb, maskb, out);
}